// VQ_64037962383654
// MI455X (gfx1250) — compile-verified
//
#include <hip/hip_runtime.h>
#include <hip/hip_bf16.h>

// VQ-VAE codebook lookup for MI455X (gfx1250, wave32, WMMA).
// Sizes fixed by the reference: N=32, C=64, H=W=64, D=512.
#define NB   32
#define CCH  64
#define HW   4096           // 64*64
#define PIX  131072         // NB*HW
#define NCD  512            // codebook entries
#define EMB_ELEMS 8388608   // NB*CCH*HW

// Channel-pair LDS row stride in float2 units.  Byte stride per row =
// 528*8 => in 4B banks: 1056 mod 64 == 32, so the two half-waves (rows q and
// q+1) land on disjoint 32-bank halves -> conflict-free ds_load_b64.
#define S2 528

typedef __attribute__((ext_vector_type(2))) float v2f;
typedef __attribute__((ext_vector_type(4))) float v4f;
typedef __attribute__((ext_vector_type(8))) float v8f;
typedef __attribute__((ext_vector_type(4))) int   v4i;

// ---------------------------------------------------------------------------
// Kernel 1: fused fp32 WMMA GEMM (Xp[16,64] x dictT[64,16-tile]) + argmin.
// 256 threads = 8 waves; each wave handles 16 pixels; block handles 128 pixels.
// Dictionary staged in LDS as channel pairs: ldsP[q][j] = {dict[j][2q], dict[j][2q+1]}.
// ---------------------------------------------------------------------------
__global__ __launch_bounds__(256)
void vq_argmin_kernel(const float* __restrict__ x,     // [NB, CCH, HW]
                      const float* __restrict__ dict,  // [NCD, CCH]
                      int* __restrict__ idx_out)       // [PIX]
{
    __shared__ v2f   ldsP[32 * S2];          // paired-channel transposed dict
    __shared__ float ldsN[NCD];              // ||d_j||^2

    const int tid   = threadIdx.x;
    const int lane  = tid & 31;
    const int wave  = tid >> 5;
    const int jl    = lane & 15;             // column within 16-wide code tile
    const int khalf = lane >> 4;             // 0: K rows {4k,4k+1}, 1: {4k+2,4k+3}

    // ---- stage dictionary into LDS (paired-transposed), coalesced reads ----
    float* ldsPf = (float*)ldsP;
    #pragma unroll 4
    for (int e = tid; e < NCD * CCH; e += 256) {
        int j = e >> 6;          // code
        int c = e & 63;          // channel
        ldsPf[((c >> 1) * S2 + j) * 2 + (c & 1)] = dict[e];
    }
    __syncthreads();

    // ---- precompute squared norms of codes ----
    for (int j = tid; j < NCD; j += 256) {
        float s = 0.0f;
        #pragma unroll
        for (int q = 0; q < 32; ++q) {
            v2f d = ldsP[q * S2 + j];
            s += d.x * d.x + d.y * d.y;
        }
        ldsN[j] = s;
    }
    __syncthreads();

    // ---- preload A matrix: 16 pixels x 64 channels into 16 float2 regs ----
    // A 16x4 fp32 layout: lanes 0-15 hold M=lane, K={0,1}; lanes 16-31 hold
    // M=lane-16, K={2,3}.  Channel loads are coalesced across the 16 pixels.
    const int pbase = (blockIdx.x * 8 + wave) * 16;
    const int mp    = pbase + jl;            // this lane's pixel row M
    const int n     = mp >> 12;              // image index
    const int hw    = mp & 4095;             // spatial offset
    const float* xb = x + ((size_t)n * CCH) * HW + hw;

    v2f a[16];
    #pragma unroll
    for (int k2 = 0; k2 < 16; ++k2) {
        int c0 = k2 * 4 + khalf * 2;
        a[k2].x = xb[(size_t)c0 * HW];
        a[k2].y = xb[(size_t)(c0 + 1) * HW];
    }

    // ---- sweep 32 code tiles: WMMA accumulate K=64, then running argmin ----
    float bestv[8];
    int   besti[8];
    #pragma unroll
    for (int v = 0; v < 8; ++v) { bestv[v] = 3.4e38f; besti[v] = 0; }

    for (int t = 0; t < 32; ++t) {
        const int j0 = t * 16;
        v8f acc = {};
        #pragma unroll
        for (int k2 = 0; k2 < 16; ++k2) {
            // One ds_load_b64 per k-step: b.x -> B VGPR0 (K=4k2 / 4k2+2),
            // b.y -> B VGPR1 (K=4k2+1 / 4k2+3), column j0+jl.
            v2f b = ldsP[(2 * k2 + khalf) * S2 + j0 + jl];
            acc = __builtin_amdgcn_wmma_f32_16x16x4_f32(
                      false, a[k2], false, b, (short)0, acc, false, false);
        }
        // score = ||d||^2 - 2*dot  (||x||^2 constant per pixel, drops out)
        float dn = ldsN[j0 + jl];
        int   j  = j0 + jl;
        #pragma unroll
        for (int v = 0; v < 8; ++v) {
            float s = dn - 2.0f * acc[v];
            if (s < bestv[v]) { bestv[v] = s; besti[v] = j; }  // strict < keeps lowest idx
        }
    }

    // ---- cross-lane min within each 16-lane half (C/D: lane%16 = column) ----
    #pragma unroll
    for (int v = 0; v < 8; ++v) {
        float bv = bestv[v];
        int   bi = besti[v];
        #pragma unroll
        for (int m = 1; m < 16; m <<= 1) {
            float ov = __shfl_xor(bv, m, 32);
            int   oi = __shfl_xor(bi, m, 32);
            if (ov < bv || (ov == bv && oi < bi)) { bv = ov; bi = oi; }
        }
        bestv[v] = bv; besti[v] = bi;
    }

    // VGPR v holds pixel row M=v (lanes 0-15) / M=v+8 (lanes 16-31)
    if (jl == 0) {
        int prow = pbase + (khalf << 3);
        #pragma unroll
        for (int v = 0; v < 8; ++v) idx_out[prow + v] = besti[v];
    }
}

// ---------------------------------------------------------------------------
// Kernel 2: embedded[n,c,h,w] = dict[idx[n,h,w], c]; embedded_pt identical.
// Pure bandwidth: vectorized idx reads + b128 nontemporal stores; dict
// gathers hit a 128 KB cache-resident footprint.
// ---------------------------------------------------------------------------
__global__ __launch_bounds__(256)
void vq_gather_kernel(const float* __restrict__ dict,
                      const int* __restrict__ idx,
                      float* __restrict__ emb,
                      float* __restrict__ emb_pt)
{
    int o  = (blockIdx.x * 256 + threadIdx.x) * 4;   // 0 .. EMB_ELEMS-1, step 4
    int hw = o & 4095;
    int c  = (o >> 12) & 63;
    int n  = o >> 18;                                // CCH*HW = 2^18
    int p  = (n << 12) | hw;

    v4i ip = *(const v4i*)(idx + p);                 // coalesced b128 load
    v4f v;
    v.x = dict[ip.x * CCH + c];
    v.y = dict[ip.y * CCH + c];
    v.z = dict[ip.z * CCH + c];
    v.w = dict[ip.w * CCH + c];
    __builtin_nontemporal_store(v, (v4f*)(emb + o));
    __builtin_nontemporal_store(v, (v4f*)(emb_pt + o));
}

extern "C" void kernel_launch(void* const* d_in, const int* in_sizes, int n_in,
                              void* d_out, int out_size, void* d_ws, size_t ws_size,
                              hipStream_t stream) {
    const float* x    = (const float*)d_in[0];   // [32,64,64,64] fp32
    const float* dict = (const float*)d_in[1];   // [512,64] fp32

    float* out    = (float*)d_out;
    float* emb    = out;                         // embedded      [32,64,64,64]
    float* emb_pt = out + (size_t)EMB_ELEMS;     // embedded_pt   (same values)
    int*   idx    = (int*)(out + (size_t)2 * EMB_ELEMS);  // idxs [32,64,64] int32

    // 131072 pixels / (8 waves * 16 pixels) = 1024 blocks
    vq_argmin_kernel<<<1024, 256, 0, stream>>>(x, dict, idx);
    // 8388608 elements / (4 per thread * 256) = 8192 blocks
    vq_gather_kernel<<<8192, 256, 0, stream>>>(dict, idx, emb, emb_pt);
}